// TwoCompartmentLIF_50826642981157
// MI455X (gfx1250) — compile-verified
//
#include <hip/hip_runtime.h>

// ---------------------------------------------------------------------------
// Two-compartment LIF scan, T=200, B=16, N=4096.  Memory-bound stream:
// ~367 MB total @ 23.3 TB/s => ~16 us floor.  One thread per neuron, 12
// state vars in VGPRs.  Inputs stream through a 3-slot LDS ring filled by
// TDM (tensor_load_to_lds) at prefetch distance 2; one block barrier per
// timestep; outputs stored non-temporal (no L2 reuse anywhere).
// ---------------------------------------------------------------------------

typedef __attribute__((ext_vector_type(4))) unsigned int u32x4;
typedef __attribute__((ext_vector_type(8))) int          i32x8;
typedef __attribute__((ext_vector_type(4))) int          i32x4;

#define T_STEPS 200
#define BATCH   16
#define NNEUR   4096
#define BN      (BATCH * NNEUR)   // 65536
#define CHUNK   256               // neurons per block == blockDim.x
#define NSLOT   3                 // LDS ring depth (prefetch distance 2)

// Issue one TDM load of `nelem` fp32 elements from global -> LDS.
// D# per CDNA5 ISA ch.8: group0 = {count=1, lds_addr, global_addr, type=2},
// group1 = {data_size=4B, tensor_dim0=tile_dim0=nelem, dim1=1, stride0=nelem},
// groups 2/3 + trailing group zero (<=2D tensor).
__device__ __forceinline__ void tdm_load_1d(const float* gptr,
                                            const float* lds_ptr,
                                            unsigned nelem) {
  unsigned long long ga  = (unsigned long long)(const void*)gptr;
  unsigned           lds = (unsigned)(unsigned long long)(const void*)lds_ptr;
  u32x4 g0;
  g0[0] = 1u;                                             // count=1, user mode
  g0[1] = lds;                                            // lds_addr (bytes)
  g0[2] = (unsigned)ga;                                   // global_addr[31:0]
  g0[3] = ((unsigned)(ga >> 32) & 0x01FFFFFFu)            // global_addr[56:32]
          | 0x80000000u;                                  // type=2 ("image")
  i32x8 g1;
  g1[0] = (int)(2u << 16);                                // data_size = 4 bytes
  g1[1] = (int)((nelem & 0xFFFFu) << 16);                 // tensor_dim0 lo16
  g1[2] = (int)((nelem >> 16) | (1u << 16));              // dim0 hi16 | dim1=1
  g1[3] = (int)(nelem << 16);                             // tile_dim0
  g1[4] = 0;                                              // tile_dim1/2 unused
  g1[5] = (int)nelem;                                     // dim0_stride lo32
  g1[6] = 0;
  g1[7] = 0;
  i32x4 gz4 = (i32x4){0, 0, 0, 0};
  i32x8 gz8 = (i32x8){0, 0, 0, 0, 0, 0, 0, 0};
  __builtin_amdgcn_tensor_load_to_lds(g0, g1, gz4, gz4, gz8, 0); // cpol = 0
}

__global__ __launch_bounds__(CHUNK)
void lif2c_scan_kernel(const float* __restrict__ exc_b,
                       const float* __restrict__ inh_b,
                       const float* __restrict__ exc_a,
                       const float* __restrict__ inh_a,
                       const float* __restrict__ noise,
                       float* __restrict__ out_spk,
                       float* __restrict__ out_v) {
  __shared__ float sb[NSLOT][5][CHUNK];

  const int tid  = threadIdx.x;
  const int base = blockIdx.x * CHUNK;

  // exp(-dt/tau) decays and exact OU sigma, precomputed in fp32.
  const float dE  = 0.81873077f;   // exp(-1/5)
  const float dI  = 0.90483743f;   // exp(-1/10)
  const float dN  = 0.99004983f;   // exp(-1/100)
  const float dCa = 0.95122945f;   // exp(-1/20)
  const float dAd = 0.99004983f;   // exp(-1/100)
  const float dOU = 0.71653131f;   // exp(-1/3)
  const float ouS = 0.05580210f;   // 0.08*sqrt(1-dOU^2)

  float Vs = 0.f, Vd = 0.f;
  float gEb = 0.f, gIb = 0.f, gNb = 0.f;
  float gEa = 0.f, gIa = 0.f, gNa = 0.f;
  float gCa = 0.f, gAd = 0.f, ref = 0.f, ou = 0.f;

  // Prologue: wave0 prefetches timesteps 0 and 1 into slots 0 and 1.
  if (tid < 32) {
    for (int p = 0; p < 2; ++p) {
      const size_t off = (size_t)p * BN + (size_t)base;
      tdm_load_1d(exc_b + off, &sb[p][0][0], CHUNK);
      tdm_load_1d(inh_b + off, &sb[p][1][0], CHUNK);
      tdm_load_1d(exc_a + off, &sb[p][2][0], CHUNK);
      tdm_load_1d(inh_a + off, &sb[p][3][0], CHUNK);
      tdm_load_1d(noise + off, &sb[p][4][0], CHUNK);
    }
  }

  int slot = 0;  // = t % NSLOT
  int pref = 2;  // = (t + 2) % NSLOT
  for (int t = 0; t < T_STEPS; ++t) {
    // Invariant at top of step t: TDM ops for steps t and t+1 (if any) are
    // in flight; in-order completion => TENSORcnt<=5 means step t is done.
    if (tid < 32) {
      if (t + 1 < T_STEPS) __builtin_amdgcn_s_wait_tensorcnt(5);
      else                 __builtin_amdgcn_s_wait_tensorcnt(0);
    }
    __syncthreads();   // publish slot `slot`; also fences prior-step readers

    const float xb = sb[slot][0][tid] * 0.02f;
    const float ib = sb[slot][1][tid] * 0.02f;
    const float xa = sb[slot][2][tid] * 0.02f;
    const float ia = sb[slot][3][tid] * 0.02f;
    const float ep = sb[slot][4][tid];

    // Conductance decays + drive (AMPA/NMDA split 0.8/0.2).
    gEb = gEb * dE + xb * 0.8f;
    gNb = gNb * dN + xb * 0.2f;
    gIb = gIb * dI + ib;
    gEa = gEa * dE + xa * 0.8f;
    gNa = gNa * dN + xa * 0.2f;
    gIa = gIa * dI + ia;
    gCa *= dCa;
    gAd *= dAd;
    ou = ou * dOU + ouS * ep;

    // Mg2+ block sigmoids (v_exp_f32 transcendental, co-executes with VALU).
    const float mgs = 1.f / (1.f + __expf(-5.f * (Vs - 0.5f)));
    const float mgd = 1.f / (1.f + __expf(-5.f * (Vd - 0.5f)));

    const float Is = 0.05f * (0.f - Vs) + gEb * (3.f - Vs)
                   + gIb * (-0.5f - Vs) + gNb * mgs * (3.f - Vs)
                   + gAd * (-0.5f - Vs) + 0.05f * (Vd - Vs);
    const float Id = 0.03f * (0.f - Vd) + gEa * (3.f - Vd)
                   + gIa * (-0.5f - Vd) + gNa * mgd * (3.f - Vd)
                   + gCa * (4.f - Vd) + 0.05f * (Vs - Vd);

    float Vsn = Vs + Is + ou;         // DT/C_M = 1
    const float Vdn = Vd + 2.f * Id;  // DT/C_D = 2

    const bool inref = ref > 0.f;
    Vsn = inref ? 0.f : Vsn;
    const bool  spike = (Vsn >= 1.f) && !inref;
    const float spk   = spike ? 1.f : 0.f;

    Vs  = spike ? 0.f : Vsn;
    ref = spike ? 5.f : fmaxf(ref - 1.f, 0.f);
    gAd += spk * 0.1f;
    Vd  = Vdn + spk * 0.3f * (4.f - Vdn);      // BAP
    gCa += (Vd >= 2.f) ? 0.3f : 0.f;           // dendritic Ca spike

    // Streaming outputs: zero reuse -> non-temporal so they don't evict
    // the input stream from L2/WGP$.
    const size_t oidx = (size_t)t * BN + (size_t)base + (size_t)tid;
    __builtin_nontemporal_store(spk, &out_spk[oidx]);
    __builtin_nontemporal_store(Vs,  &out_v[oidx]);

    // Refill slot `pref` for step t+2.  Its previous readers ran at step
    // t-1 and already passed this step's barrier, so no trailing barrier
    // is needed (single barrier per timestep).
    if ((tid < 32) && (t + 2 < T_STEPS)) {
      const size_t off = (size_t)(t + 2) * BN + (size_t)base;
      tdm_load_1d(exc_b + off, &sb[pref][0][0], CHUNK);
      tdm_load_1d(inh_b + off, &sb[pref][1][0], CHUNK);
      tdm_load_1d(exc_a + off, &sb[pref][2][0], CHUNK);
      tdm_load_1d(inh_a + off, &sb[pref][3][0], CHUNK);
      tdm_load_1d(noise + off, &sb[pref][4][0], CHUNK);
    }

    slot = (slot == NSLOT - 1) ? 0 : slot + 1;
    pref = (pref == NSLOT - 1) ? 0 : pref + 1;
  }
}

extern "C" void kernel_launch(void* const* d_in, const int* in_sizes, int n_in,
                              void* d_out, int out_size, void* d_ws,
                              size_t ws_size, hipStream_t stream) {
  (void)in_sizes; (void)n_in; (void)out_size; (void)d_ws; (void)ws_size;
  const float* exc_b = (const float*)d_in[0];
  const float* inh_b = (const float*)d_in[1];
  const float* exc_a = (const float*)d_in[2];
  const float* inh_a = (const float*)d_in[3];
  const float* noise = (const float*)d_in[4];
  float* out_spk = (float*)d_out;                         // [T,B,N]
  float* out_v   = out_spk + (size_t)T_STEPS * BN;        // [T,B,N]

  dim3 grid(BN / CHUNK);   // 256 blocks
  dim3 block(CHUNK);       // 256 threads = 8 wave32 waves
  hipLaunchKernelGGL(lif2c_scan_kernel, grid, block, 0, stream,
                     exc_b, inh_b, exc_a, inh_a, noise, out_spk, out_v);
}